// DGCNN_ref_43482248905371
// MI455X (gfx1250) — compile-verified
//
#include <hip/hip_runtime.h>
#include <hip/hip_bf16.h>

// ---------------------------------------------------------------------------
// DGCNN forward + loss for MI455X (gfx1250, wave32, WMMA).
//
// EdgeConv algebraic decomposition:
//   conv(concat(nbr-ctr, ctr)) = P[:, idx] + Q[:, n],
//   P = W1 @ X, Q = (W2-W1) @ X      -> two small GEMMs + gather
//   max_k lrelu(bn(t)) = max(f(max_k t), f(min_k t))  (affine f, any sign)
//
// All GEMMs use V_WMMA_F32_16X16X4_F32 (f32 A/B, f32 acc) so KNN ordering and
// activations track the f32 reference. K is a template parameter (4/64/128/512,
// layer-1 input zero-padded 3->4 channels) so inner loops are branch-free.
// GEMM blocks stage the shared [K x 16] B panel in LDS via
// global_load_async_to_lds_b32 (ASYNCcnt) when the builtin is available.
// ---------------------------------------------------------------------------

typedef float v2f __attribute__((ext_vector_type(2)));
typedef float v8f __attribute__((ext_vector_type(8)));

#define NPTS    2048
#define BATCH   8
#define KNN     20
#define LRELU_S 0.2f
#define BN_EPSF 1e-5f

#if defined(__HIP_DEVICE_COMPILE__) && defined(__gfx1250__) && \
    __has_builtin(__builtin_amdgcn_global_load_async_to_lds_b32)
#define HAVE_ASYNC_LDS 1
#else
#define HAVE_ASYNC_LDS 0
#endif

__device__ __forceinline__ v8f wmma_f32_16x16x4(v2f a, v2f b, v8f c) {
  // 8 args: (neg_a, A, neg_b, B, c_mod, C, reuse_a, reuse_b)
  return __builtin_amdgcn_wmma_f32_16x16x4_f32(false, a, false, b, (short)0, c,
                                               false, false);
}

__device__ __forceinline__ float lrelu(float x) {
  return x >= 0.f ? x : LRELU_S * x;
}

__device__ __forceinline__ void async_copy_b32(const float* src, float* dst_lds) {
#if HAVE_ASYNC_LDS
  __builtin_amdgcn_global_load_async_to_lds_b32(
      (__attribute__((address_space(1))) int*)(unsigned long long)src,
      (__attribute__((address_space(3))) int*)(unsigned long long)dst_lds, 0, 0);
#else
  *dst_lds = *src;
#endif
}

__device__ __forceinline__ void async_copy_wait() {
#if HAVE_ASYNC_LDS
#if __has_builtin(__builtin_amdgcn_s_wait_asynccnt)
  __builtin_amdgcn_s_wait_asynccnt(0);
#else
  asm volatile("s_wait_asynccnt 0" ::: "memory");
#endif
#endif
}

// ---------------------------------------------------------------------------
// Zero-pad x [B][3][2048] -> Xpad [B][4][2048] (row 3 = 0) so layer-1 K=4.
// ---------------------------------------------------------------------------
__global__ __launch_bounds__(256) void pad_x_kernel(const float* __restrict__ x,
                                                    float* __restrict__ Xpad) {
  const int gid = blockIdx.x * 256 + threadIdx.x;  // over B*4*2048
  const int b = gid >> 13;                         // / (4*2048)
  const int r = (gid >> 11) & 3;
  const int n = gid & (NPTS - 1);
  Xpad[gid] = (r < 3) ? x[((size_t)b * 3 + r) * NPTS + n] : 0.f;
}

// ---------------------------------------------------------------------------
// Per-point squared norms: norms[b][n] = sum_c X[c][n]^2
// ---------------------------------------------------------------------------
__global__ __launch_bounds__(256) void norms_kernel(const float* __restrict__ Xall,
                                                    long long sX, int C,
                                                    float* __restrict__ norms) {
  const int b = blockIdx.y;
  const int n = blockIdx.x * 256 + threadIdx.x;
  const float* X = Xall + (size_t)b * sX;
  float s = 0.f;
  for (int c = 0; c < C; ++c) {
    float t = X[(size_t)c * NPTS + n];
    s += t * t;
  }
  norms[(size_t)b * NPTS + n] = s;
}

// ---------------------------------------------------------------------------
// KNN: one block per 16-row tile; 8 waves sweep 16x16 column tiles with
// V_WMMA_F32_16X16X4_F32 (Gram matrix). A fragments (this row tile) are
// loop-invariant -> hoisted to registers. pdist stripe lives in 128KB LDS,
// then iterative top-20 per row (first-occurrence ties, as lax.top_k).
// ---------------------------------------------------------------------------
template <int K>
__global__ __launch_bounds__(256) void knn_topk_kernel(const float* __restrict__ Xall,
                                                       long long sX,
                                                       const float* __restrict__ norms,
                                                       int* __restrict__ idxOut) {
  extern __shared__ float sdist[];  // 16 * 2048 floats = 128 KB
  const int b    = blockIdx.y;
  const int row0 = blockIdx.x * 16;
  const int wave = threadIdx.x >> 5;
  const int lane = threadIdx.x & 31;
  const int l    = lane & 15;
  const int half = lane >> 4;
  const float* X  = Xall  + (size_t)b * sX;
  const float* nr = norms + (size_t)b * NPTS;

  // Hoisted A fragments: A[m][k] = X[k][row0+m]; lane half selects K pair.
  v2f afr[K / 4];
#pragma unroll
  for (int kk = 0; kk < K; kk += 4) {
    const int k0 = kk + 2 * half;
    afr[kk >> 2].x = X[(size_t)k0 * NPTS + row0 + l];
    afr[kk >> 2].y = X[(size_t)(k0 + 1) * NPTS + row0 + l];
  }

  for (int ct = wave; ct < NPTS / 16; ct += 8) {
    const int col0 = ct * 16;
    v8f acc = {0.f, 0.f, 0.f, 0.f, 0.f, 0.f, 0.f, 0.f};
#pragma unroll
    for (int kk = 0; kk < K; kk += 4) {
      const int k0 = kk + 2 * half;
      v2f bv;
      bv.x = X[(size_t)k0 * NPTS + col0 + l];
      bv.y = X[(size_t)(k0 + 1) * NPTS + col0 + l];
      acc = wmma_f32_16x16x4(afr[kk >> 2], bv, acc);
    }
    const float cn = nr[col0 + l];
#pragma unroll
    for (int v = 0; v < 8; ++v) {
      const int r = v + 8 * half;  // C/D layout: vgpr v + half -> row; lane&15 -> col
      sdist[r * NPTS + col0 + l] = 2.f * acc[v] - nr[row0 + r] - cn;
    }
  }
  __syncthreads();

  // selection: 2 rows per wave, 20 iterative max picks each
  for (int rr = 0; rr < 2; ++rr) {
    const int r = wave * 2 + rr;
    volatile float* sd = sdist + r * NPTS;
    int* op = idxOut + ((size_t)b * NPTS + row0 + r) * KNN;
    for (int kk = 0; kk < KNN; ++kk) {
      float best = -3.4e38f;
      int   bi   = 0x7fffffff;
      for (int j = lane; j < NPTS; j += 32) {
        float vv = sd[j];
        if (vv > best || (vv == best && j < bi)) { best = vv; bi = j; }
      }
#pragma unroll
      for (int o = 16; o > 0; o >>= 1) {
        float ov = __shfl_xor(best, o, 32);
        int   oi = __shfl_xor(bi, o, 32);
        if (ov > best || (ov == best && oi < bi)) { best = ov; bi = oi; }
      }
      if (lane == 0) { op[kk] = bi; sd[bi] = -3.4e38f; }
      // wave32 lockstep + in-order DS ops: store visible to next scan
    }
  }
}

// ---------------------------------------------------------------------------
// Build Wcat = [W1 ; W2-W1] padded to Cpad columns (zeros beyond C).
// w is [O][2C] row-major; W1 = w[:, :C], W2 = w[:, C:].
// ---------------------------------------------------------------------------
__global__ __launch_bounds__(256) void prep_wcat_kernel(const float* __restrict__ w,
                                                        float* __restrict__ wcat,
                                                        int O, int C, int Cpad) {
  const int gid = blockIdx.x * 256 + threadIdx.x;
  const int total = 2 * O * Cpad;
  if (gid >= total) return;
  const int r = gid / Cpad;
  const int c = gid % Cpad;
  float val = 0.f;
  if (c < C) {
    if (r < O) {
      val = w[(size_t)r * (2 * C) + c];
    } else {
      const int o = r - O;
      val = w[(size_t)o * (2 * C) + C + c] - w[(size_t)o * (2 * C) + c];
    }
  }
  wcat[gid] = val;
}

// ---------------------------------------------------------------------------
// Panel GEMM: D[b] = A(MxK) * Bmat[b](K x 2048).
// Block = 256 threads (8 waves) covering 128 output rows x one 16-col panel.
// The shared [K x 16] B panel is staged once into LDS (async-to-LDS when
// available), pair-interleaved so compute reads v2f via ds_load_b64.
// Optional fused BN + LeakyReLU epilogue (conv5).
// ---------------------------------------------------------------------------
template <int K>
__global__ __launch_bounds__(256) void gemm_panel_kernel(
    const float* __restrict__ A, int lda,
    const float* __restrict__ Bmat, long long sB,
    float* __restrict__ D, long long sD,
    const float* __restrict__ g, const float* __restrict__ bb,
    const float* __restrict__ mm, const float* __restrict__ vv, int fuse) {
  extern __shared__ float Bs[];  // (K/2) pairs x 16 cols x 2 = K*16 floats
  const int n0   = blockIdx.x * 16;
  const int b    = blockIdx.z;
  const int wave = threadIdx.x >> 5;
  const int lane = threadIdx.x & 31;
  const int l    = lane & 15;
  const int half = lane >> 4;
  const int m0   = blockIdx.y * 128 + wave * 16;
  const float* Bp = Bmat + (size_t)b * sB + n0;

  // Cooperative stage: Bs[pair p][col c] = {B[2p][c], B[2p+1][c]}
  {
    const int c  = threadIdx.x & 15;
    const int kb = threadIdx.x >> 4;  // 0..15
#pragma unroll
    for (int k = kb; k < K; k += 16) {
      const int p = k >> 1;
      async_copy_b32(Bp + (size_t)k * NPTS + c, Bs + p * 32 + c * 2 + (k & 1));
    }
    async_copy_wait();
  }
  __syncthreads();

  v8f acc = {0.f, 0.f, 0.f, 0.f, 0.f, 0.f, 0.f, 0.f};
  const float* Ar = A + (size_t)(m0 + l) * lda;
#pragma unroll 8
  for (int kk = 0; kk < K; kk += 4) {
    const int k0 = kk + 2 * half;
    const v2f av = *(const v2f*)(Ar + k0);                       // global_load_b64
    const v2f bv = *(const v2f*)(Bs + ((kk >> 1) + half) * 32 + l * 2);  // ds_load_b64
    acc = wmma_f32_16x16x4(av, bv, acc);
  }

  float* Dp = D + (size_t)b * sD;
#pragma unroll
  for (int v = 0; v < 8; ++v) {
    const int r = m0 + v + 8 * half;
    float val = acc[v];
    if (fuse) {
      const float a = g[r] * rsqrtf(vv[r] + BN_EPSF);
      val = a * (val - mm[r]) + bb[r];
      val = lrelu(val);
    }
    Dp[(size_t)r * NPTS + n0 + l] = val;
  }
}

// ---------------------------------------------------------------------------
// EdgeConv aggregation: out[o][n] = max_k lrelu(bn(P[o,idx[n,k]] + Q[o,n]))
// via max/min of P over neighbors (affine-monotone trick).
// PQ per batch: rows [0,O) = P, rows [O,2O) = Q.
// ---------------------------------------------------------------------------
__global__ __launch_bounds__(256) void aggregate_kernel(
    const float* __restrict__ PQ, const int* __restrict__ idx,
    const float* __restrict__ g, const float* __restrict__ bb,
    const float* __restrict__ mm, const float* __restrict__ vv,
    float* __restrict__ out, int O, long long sPQ, long long sOut) {
  const int b   = blockIdx.y;
  const int gid = blockIdx.x * 256 + threadIdx.x;
  const int o   = gid >> 11;  // / 2048
  const int n   = gid & (NPTS - 1);
  const float* P = PQ + (size_t)b * sPQ + (size_t)o * NPTS;
  const float* Q = PQ + (size_t)b * sPQ + (size_t)(O + o) * NPTS;
  const int* id  = idx + ((size_t)b * NPTS + n) * KNN;
  float tmax = -3.4e38f, tmin = 3.4e38f;
#pragma unroll
  for (int k = 0; k < KNN; ++k) {
    const float p = P[id[k]];
    tmax = fmaxf(tmax, p);
    tmin = fminf(tmin, p);
  }
  const float q = Q[n];
  const float a = g[o] * rsqrtf(vv[o] + BN_EPSF);
  const float c = bb[o] - a * mm[o];
  const float y1 = lrelu(a * (tmax + q) + c);
  const float y2 = lrelu(a * (tmin + q) + c);
  out[(size_t)b * sOut + (size_t)o * NPTS + n] = fmaxf(y1, y2);
}

// ---------------------------------------------------------------------------
// Row max (+ first-occurrence argmax) over 2048 points; one wave per (b,o).
// mode 1: mask points with used[b][n]!=0 to NEG_INF (-1e30, as reference).
// ---------------------------------------------------------------------------
__global__ __launch_bounds__(256) void rowmax_kernel(const float* __restrict__ X5,
                                                     float* __restrict__ feat,
                                                     int* __restrict__ argm,
                                                     const int* __restrict__ used,
                                                     int mode) {
  const int wave = threadIdx.x >> 5;
  const int lane = threadIdx.x & 31;
  const int w    = blockIdx.x * 8 + wave;  // 0..8191
  const int b    = w >> 10;
  const int o    = w & 1023;
  const float* row = X5 + ((size_t)b * 1024 + o) * NPTS;
  float best = -3.4e38f;
  int   bi   = 0x7fffffff;
  for (int j = lane; j < NPTS; j += 32) {
    float vv = row[j];
    if (mode == 1 && used[b * NPTS + j]) vv = -1e30f;
    if (vv > best || (vv == best && j < bi)) { best = vv; bi = j; }
  }
#pragma unroll
  for (int o2 = 16; o2 > 0; o2 >>= 1) {
    float ov = __shfl_xor(best, o2, 32);
    int   oi = __shfl_xor(bi, o2, 32);
    if (ov > best || (ov == best && oi < bi)) { best = ov; bi = oi; }
  }
  if (lane == 0) {
    feat[b * 1024 + o] = best;
    if (argm) argm[b * 1024 + o] = bi;
  }
}

__global__ __launch_bounds__(256) void clear_int_kernel(int* __restrict__ p, int n) {
  const int gid = blockIdx.x * 256 + threadIdx.x;
  if (gid < n) p[gid] = 0;
}

__global__ __launch_bounds__(256) void scatter_used_kernel(const int* __restrict__ idx1,
                                                           int* __restrict__ used) {
  const int gid = blockIdx.x * 256 + threadIdx.x;
  if (gid >= BATCH * 1024) return;
  const int b = gid >> 10;
  used[b * NPTS + idx1[gid]] = 1;  // all writers store 1: race is benign
}

// ---------------------------------------------------------------------------
// Fully-connected: out[b][j] = act(bn(in[b]·W[j] + bias)) — small sizes.
// ---------------------------------------------------------------------------
__global__ __launch_bounds__(256) void fc_kernel(const float* __restrict__ in, int Cin,
                                                 const float* __restrict__ W,
                                                 const float* __restrict__ bias,
                                                 const float* __restrict__ g,
                                                 const float* __restrict__ bb,
                                                 const float* __restrict__ mm,
                                                 const float* __restrict__ vv,
                                                 int act, int Cout,
                                                 float* __restrict__ out) {
  const int gid = blockIdx.x * 256 + threadIdx.x;
  if (gid >= BATCH * Cout) return;
  const int b = gid / Cout;
  const int j = gid % Cout;
  const float* ir = in + (size_t)b * Cin;
  const float* wr = W + (size_t)j * Cin;
  float s = 0.f;
  for (int i = 0; i < Cin; ++i) s += ir[i] * wr[i];
  if (bias) s += bias[j];
  if (g) {
    const float a = g[j] * rsqrtf(vv[j] + BN_EPSF);
    s = a * (s - mm[j]) + bb[j];
  }
  if (act) s = lrelu(s);
  out[gid] = s;
}

// ---------------------------------------------------------------------------
// Loss: smooth CE on pred1/pred2 + aug term. Writes pred1 (320) + loss (1).
// ---------------------------------------------------------------------------
__global__ __launch_bounds__(64) void loss_kernel(const float* __restrict__ p1,
                                                  const float* __restrict__ p2,
                                                  const int* __restrict__ y,
                                                  float* __restrict__ out) {
  __shared__ float l1s[BATCH], l2s[BATCH], augs[BATCH];
  const int t = threadIdx.x;
  const int NC = 40;
  if (t < BATCH) {
    const float* a = p1 + t * NC;
    const float* c = p2 + t * NC;
    const int yy = y[t];
    float m1 = -3.4e38f, m2 = -3.4e38f;
    for (int i = 0; i < NC; ++i) { m1 = fmaxf(m1, a[i]); m2 = fmaxf(m2, c[i]); }
    float s1 = 0.f, s2 = 0.f, sa = 0.f, sc = 0.f;
    for (int i = 0; i < NC; ++i) {
      s1 += expf(a[i] - m1);
      s2 += expf(c[i] - m2);
      sa += a[i];
      sc += c[i];
    }
    const float lse1 = m1 + logf(s1);
    const float lse2 = m2 + logf(s2);
    const float eps = 0.2f, offc = eps / (NC - 1);
    const float d1 = (1.f - eps) * a[yy] + offc * (sa - a[yy]);
    const float d2 = (1.f - eps) * c[yy] + offc * (sc - c[yy]);
    const float l1r = lse1 - d1;  // sum(oh)=1 -> lse - oh·p
    const float l2r = lse2 - d2;
    const float pc_con = expf(a[yy] - lse1);
    const float par = fmaxf(2.1f, expf(pc_con) * 2.1f);
    l1s[t] = l1r;
    l2s[t] = l2r;
    augs[t] = fabsf(1.f - expf(l2r - l1r * par));
  }
  __syncthreads();
  if (t == 0) {
    float L = 0.f;
    for (int b = 0; b < BATCH; ++b) L += (l1s[b] + l2s[b] + augs[b]) * (1.f / BATCH);
    out[BATCH * 40] = L;
  }
  for (int i = t; i < BATCH * 40; i += blockDim.x) out[i] = p1[i];
}

// ---------------------------------------------------------------------------
// Host orchestration. Input order (top-level insertion order; params dict in
// insertion order, bn tuples expanded g,b,m,v):
//   0:x 1:c1_w 2-5:bn1 6:c2_w 7-10:bn2 11:c3_w 12-15:bn3 16:c4_w 17-20:bn4
//   21:c5_w 22-25:bn5 26:l1_w 27-30:bn6 31:l2_w 32:l2_b 33-36:bn7 37:l3_w
//   38:l3_b 39:y
// ---------------------------------------------------------------------------
extern "C" void kernel_launch(void* const* d_in, const int* in_sizes, int n_in,
                              void* d_out, int out_size, void* d_ws, size_t ws_size,
                              hipStream_t stream) {
  (void)in_sizes; (void)out_size;
  if (!d_in || n_in < 40 || !d_out || !d_ws) return;

  const float* x = (const float*)d_in[0];
  const float* c1_w = (const float*)d_in[1];
  const float* bn1[4] = {(const float*)d_in[2], (const float*)d_in[3],
                         (const float*)d_in[4], (const float*)d_in[5]};
  const float* c2_w = (const float*)d_in[6];
  const float* bn2[4] = {(const float*)d_in[7], (const float*)d_in[8],
                         (const float*)d_in[9], (const float*)d_in[10]};
  const float* c3_w = (const float*)d_in[11];
  const float* bn3[4] = {(const float*)d_in[12], (const float*)d_in[13],
                         (const float*)d_in[14], (const float*)d_in[15]};
  const float* c4_w = (const float*)d_in[16];
  const float* bn4[4] = {(const float*)d_in[17], (const float*)d_in[18],
                         (const float*)d_in[19], (const float*)d_in[20]};
  const float* c5_w = (const float*)d_in[21];
  const float* bn5[4] = {(const float*)d_in[22], (const float*)d_in[23],
                         (const float*)d_in[24], (const float*)d_in[25]};
  const float* l1_w = (const float*)d_in[26];
  const float* bn6[4] = {(const float*)d_in[27], (const float*)d_in[28],
                         (const float*)d_in[29], (const float*)d_in[30]};
  const float* l2_w = (const float*)d_in[31];
  const float* l2_b = (const float*)d_in[32];
  const float* bn7[4] = {(const float*)d_in[33], (const float*)d_in[34],
                         (const float*)d_in[35], (const float*)d_in[36]};
  const float* l3_w = (const float*)d_in[37];
  const float* l3_b = (const float*)d_in[38];
  const int* y = (const int*)d_in[39];

  // ---- workspace carve ----
  char* base = (char*)d_ws;
  size_t off = 0;
  auto take = [&](size_t bytes) -> void* {
    off = (off + 255) & ~(size_t)255;
    void* p = base + off;
    off += bytes;
    return p;
  };
  float* Xpad  = (float*)take((size_t)BATCH * 4 * NPTS * 4);
  float* norms = (float*)take((size_t)BATCH * NPTS * 4);
  int*   idx   = (int*)take((size_t)BATCH * NPTS * KNN * 4);
  float* wcat  = (float*)take((size_t)512 * 128 * 4);
  float* PQ    = (float*)take((size_t)BATCH * 512 * NPTS * 4);
  float* Xcat  = (float*)take((size_t)BATCH * 512 * NPTS * 4);
  float* X5    = (float*)take((size_t)BATCH * 1024 * NPTS * 4);
  float* feat1 = (float*)take((size_t)BATCH * 1024 * 4);
  float* feat2 = (float*)take((size_t)BATCH * 1024 * 4);
  int*   idx1  = (int*)take((size_t)BATCH * 1024 * 4);
  int*   used  = (int*)take((size_t)BATCH * NPTS * 4);
  float* h1    = (float*)take((size_t)BATCH * 512 * 4);
  float* h2    = (float*)take((size_t)BATCH * 256 * 4);
  float* p1    = (float*)take((size_t)BATCH * 40 * 4);
  float* p2    = (float*)take((size_t)BATCH * 40 * 4);
  if (off > ws_size) return;  // insufficient scratch; do nothing (deterministic)

  const long long sXcat = (long long)512 * NPTS;

  pad_x_kernel<<<(BATCH * 4 * NPTS) / 256, 256, 0, stream>>>(x, Xpad);

  struct Layer {
    const float* Xin; long long sX; int Ck /*GEMM K*/, Cw /*weight C*/, O;
    const float* w; const float* const* bn; int outRow;
  };
  const Layer L[4] = {
      {Xpad,              (long long)4 * NPTS, 4,   3,   64,  c1_w, bn1, 0},
      {Xcat,              sXcat,               64,  64,  64,  c2_w, bn2, 64},
      {Xcat + 64 * NPTS,  sXcat,               64,  64,  128, c3_w, bn3, 128},
      {Xcat + 128 * NPTS, sXcat,               128, 128, 256, c4_w, bn4, 256},
  };

  for (int li = 0; li < 4; ++li) {
    const Layer& P = L[li];
    norms_kernel<<<dim3(NPTS / 256, BATCH), 256, 0, stream>>>(P.Xin, P.sX, P.Ck, norms);

    const dim3 kg(NPTS / 16, BATCH);
    const size_t kshm = 16 * NPTS * sizeof(float);
    switch (P.Ck) {
      case 4:   knn_topk_kernel<4><<<kg, 256, kshm, stream>>>(P.Xin, P.sX, norms, idx); break;
      case 64:  knn_topk_kernel<64><<<kg, 256, kshm, stream>>>(P.Xin, P.sX, norms, idx); break;
      default:  knn_topk_kernel<128><<<kg, 256, kshm, stream>>>(P.Xin, P.sX, norms, idx); break;
    }

    const int twoO = 2 * P.O;
    const int tot = twoO * P.Ck;
    prep_wcat_kernel<<<(tot + 255) / 256, 256, 0, stream>>>(P.w, wcat, P.O, P.Cw, P.Ck);

    const dim3 gg(NPTS / 16, twoO / 128, BATCH);
    const size_t gshm = (size_t)P.Ck * 16 * sizeof(float);
    const long long sPQ = (long long)twoO * NPTS;
    switch (P.Ck) {
      case 4:
        gemm_panel_kernel<4><<<gg, 256, gshm, stream>>>(
            wcat, P.Ck, P.Xin, P.sX, PQ, sPQ, nullptr, nullptr, nullptr, nullptr, 0);
        break;
      case 64:
        gemm_panel_kernel<64><<<gg, 256, gshm, stream>>>(
            wcat, P.Ck, P.Xin, P.sX, PQ, sPQ, nullptr, nullptr, nullptr, nullptr, 0);
        break;
      default:
        gemm_panel_kernel<128><<<gg, 256, gshm, stream>>>(
            wcat, P.Ck, P.Xin, P.sX, PQ, sPQ, nullptr, nullptr, nullptr, nullptr, 0);
        break;
    }

    aggregate_kernel<<<dim3(P.O * NPTS / 256, BATCH), 256, 0, stream>>>(
        PQ, idx, P.bn[0], P.bn[1], P.bn[2], P.bn[3],
        Xcat + (size_t)P.outRow * NPTS, P.O, sPQ, sXcat);
  }

  // conv5: [1024,512] x Xcat, fused bn5 + lrelu
  gemm_panel_kernel<512><<<dim3(NPTS / 16, 1024 / 128, BATCH), 256,
                           (size_t)512 * 16 * sizeof(float), stream>>>(
      c5_w, 512, Xcat, sXcat, X5, (long long)1024 * NPTS,
      bn5[0], bn5[1], bn5[2], bn5[3], 1);

  // global max + refinement
  rowmax_kernel<<<BATCH * 1024 / 8, 256, 0, stream>>>(X5, feat1, idx1, nullptr, 0);
  clear_int_kernel<<<(BATCH * NPTS + 255) / 256, 256, 0, stream>>>(used, BATCH * NPTS);
  scatter_used_kernel<<<(BATCH * 1024 + 255) / 256, 256, 0, stream>>>(idx1, used);
  rowmax_kernel<<<BATCH * 1024 / 8, 256, 0, stream>>>(X5, feat2, nullptr, used, 1);

  // heads (dropout is identity in eval)
  fc_kernel<<<(BATCH * 512 + 255) / 256, 256, 0, stream>>>(
      feat1, 1024, l1_w, nullptr, bn6[0], bn6[1], bn6[2], bn6[3], 1, 512, h1);
  fc_kernel<<<(BATCH * 256 + 255) / 256, 256, 0, stream>>>(
      h1, 512, l2_w, l2_b, bn7[0], bn7[1], bn7[2], bn7[3], 1, 256, h2);
  fc_kernel<<<(BATCH * 40 + 255) / 256, 256, 0, stream>>>(
      h2, 256, l3_w, l3_b, nullptr, nullptr, nullptr, nullptr, 0, 40, p1);

  fc_kernel<<<(BATCH * 512 + 255) / 256, 256, 0, stream>>>(
      feat2, 1024, l1_w, nullptr, bn6[0], bn6[1], bn6[2], bn6[3], 1, 512, h1);
  fc_kernel<<<(BATCH * 256 + 255) / 256, 256, 0, stream>>>(
      h1, 512, l2_w, l2_b, bn7[0], bn7[1], bn7[2], bn7[3], 1, 256, h2);
  fc_kernel<<<(BATCH * 40 + 255) / 256, 256, 0, stream>>>(
      h2, 256, l3_w, l3_b, nullptr, nullptr, nullptr, nullptr, 0, 40, p2);

  loss_kernel<<<1, 64, 0, stream>>>(p1, p2, y, (float*)d_out);
}